// dense_warp_1580547973632
// MI455X (gfx1250) — compile-verified
//
#include <hip/hip_runtime.h>

typedef float v2f __attribute__((ext_vector_type(2)));
typedef float v8f __attribute__((ext_vector_type(8)));

#define B_ 4
#define C_ 32
#define H_ 256
#define W_ 512
#define D_ 48

#define COLS 128              // output columns per workgroup
#define A_COLS 192            // h1 columns staged (COLS + 64 halo)
#define H1S 196               // h1 LDS row stride (conflict-free b64 reads)
#define DSLOTS 84             // costT per-column stride; dslot = d + 15
                              // valid data d in [0,48) -> dslots 15..62
                              // guard zeros: dslots 0..14 and 63..83

__global__ __launch_bounds__(256) void banded_corr_wmma(
    const float* __restrict__ h1, const float* __restrict__ cost,
    float* __restrict__ out)
{
    // column-major skewed cost tile: costT[c][dslot], c = w - w0c
    __shared__ __attribute__((aligned(16))) float costT[COLS * DSLOTS]; // 42 KiB
    __shared__ __attribute__((aligned(16))) float h1s[C_ * H1S];        // 24.5 KiB

    const int chunk = blockIdx.x;   // 0..3
    const int h     = blockIdx.y;   // 0..255
    const int b     = blockIdx.z;   // 0..3
    const int w0c   = chunk * COLS;
    const int t     = threadIdx.x;

    // ---- zero the guard bands of costT (36 rows x 128 cols, disjoint from fill)
    for (int idx = t; idx < 36 * COLS; idx += 256) {
        int zr    = idx >> 7;                    // 0..35
        int c     = idx & (COLS - 1);
        int dslot = (zr < 15) ? zr : (zr + 48);  // 0..14 and 63..83
        costT[c * DSLOTS + dslot] = 0.0f;
    }
    // ---- fill band: costT[c][d+15] = cost[b,d,h,w0c+c], masked by (w0c+c+d < W)
    {
        const float* src = cost + ((size_t)b * D_ * H_ + h) * W_;
        for (int idx = t; idx < D_ * COLS; idx += 256) {
            int d = idx >> 7;                    // 0..47
            int c = idx & (COLS - 1);
            float v = src[(size_t)d * H_ * W_ + (w0c + c)];   // always in-bounds
            costT[c * DSLOTS + (d + 15)] = ((w0c + c + d) < W_) ? v : 0.0f;
        }
    }
    // ---- stage h1[b, 0:32, h, w0c:w0c+192] -> LDS, zero past W (branch-free)
    {
        const float* src = h1 + ((size_t)b * C_ * H_ + h) * W_;
        for (int idx = t; idx < C_ * A_COLS; idx += 256) {
            int r = idx / A_COLS;
            int c = idx - r * A_COLS;
            int w = w0c + c;
            int wcl = (w < W_) ? w : (W_ - 1);               // safe address
            float v = src[(size_t)r * H_ * W_ + wcl];
            h1s[r * H1S + c] = (w < W_) ? v : 0.0f;
        }
    }
    __syncthreads();

    // ---- per-wave 32x16 output (two 16x16 tiles); guard-free banded GEMM, K-extent 64 ---
    const int wave = t >> 5;        // 0..7 (wave32)
    const int lane = t & 31;
    const int lo   = lane & 15;     // A: M row | B: N col | C/D: N col
    const int hi   = lane >> 4;     // selects K pair / upper C/D rows
    const int wl   = wave * 16;     // tile base column within chunk

    // B element (kg = 4k+2*hi+j, n = lo) -> costT[(wl+lo)][kg - lo + 15]
    const float* bbase  = &costT[(wl + lo) * DSLOTS + (2 * hi - lo + 15)];
    // A element (m row = c16+lo, kg)     -> h1s[row][wl + kg]
    const float* abase0 = &h1s[(0  + lo) * H1S + wl + 2 * hi];
    const float* abase1 = &h1s[(16 + lo) * H1S + wl + 2 * hi];

    v8f acc0 = {0.f,0.f,0.f,0.f,0.f,0.f,0.f,0.f};
    v8f acc1 = {0.f,0.f,0.f,0.f,0.f,0.f,0.f,0.f};

    #pragma unroll
    for (int k = 0; k < 16; ++k) {
        v2f bf;
        bf.x = bbase[4 * k];
        bf.y = bbase[4 * k + 1];
        v2f a0 = *(const v2f*)(abase0 + 4 * k);   // 8B-aligned LDS b64
        v2f a1 = *(const v2f*)(abase1 + 4 * k);

        acc0 = __builtin_amdgcn_wmma_f32_16x16x4_f32(false, a0, false, bf,
                                                     (short)0, acc0, false, false);
        acc1 = __builtin_amdgcn_wmma_f32_16x16x4_f32(false, a1, false, bf,
                                                     (short)0, acc1, false, false);
    }

    // ---- write D tiles: row = 16*m + 8*hi + r, col = w0 + lo ---------------
    float* dst = out + ((size_t)b * C_ * H_ + h) * W_;
    const int wcol = w0c + wl + lo;
    #pragma unroll
    for (int r = 0; r < 8; ++r) {
        dst[(size_t)(0  + 8 * hi + r) * H_ * W_ + wcol] = acc0[r];
        dst[(size_t)(16 + 8 * hi + r) * H_ * W_ + wcol] = acc1[r];
    }
}

extern "C" void kernel_launch(void* const* d_in, const int* in_sizes, int n_in,
                              void* d_out, int out_size, void* d_ws, size_t ws_size,
                              hipStream_t stream) {
    const float* h1   = (const float*)d_in[0];
    const float* cost = (const float*)d_in[1];
    float* out        = (float*)d_out;

    dim3 grid(W_ / COLS, H_, B_);   // (4, 256, 4)
    banded_corr_wmma<<<grid, 256, 0, stream>>>(h1, cost, out);
}